// EfficientNonSamplingFactorizationMachines_72000831750869
// MI455X (gfx1250) — compile-verified
//
#include <hip/hip_runtime.h>

typedef __attribute__((ext_vector_type(2))) float v2f;
typedef __attribute__((ext_vector_type(8))) float v8f;

#define LFEAT 32
#define DIM   64
#define WAVES 16   // entities per block, one wave each

// One wave per entity: gather-reduce L=32 embedding rows (float2/lane,
// fully coalesced 256B per row), build bi in LDS, then wave 0 computes
// scal[0:16] = bi[16x64] @ h1 with 16 x V_WMMA_F32_16X16X4_F32.
__global__ __launch_bounds__(512) void enfm_entity_kernel(
    const float* __restrict__ emb,      // [V, 64]
    const float* __restrict__ w,        // [V]
    const float* __restrict__ h1,       // [64]
    const float* __restrict__ bias_ptr, // [1] or nullptr
    const int*   __restrict__ idx,      // [N, 32]
    float*       __restrict__ outRows,  // [N, 66]
    int scal_col, int ones_col, int n_entities)
{
    __shared__ float lds_bi[WAVES][DIM + 2];  // +2 pad: stagger LDS banks
    __shared__ float lds_scal[WAVES];

    const int wave = threadIdx.x >> 5;
    const int lane = threadIdx.x & 31;
    const int entity = blockIdx.x * WAVES + wave;

    float2 sum = {0.f, 0.f};
    float2 sq  = {0.f, 0.f};
    float  wpart = 0.f;

    if (entity < n_entities) {
        // one coalesced load of all 32 indices for this entity
        int my_idx = idx[(size_t)entity * LFEAT + lane];
        wpart = w[my_idx];
        #pragma unroll 4
        for (int l = 0; l < LFEAT; ++l) {
            int row = __shfl(my_idx, l, 32);   // wave-uniform row id
            float2 e = ((const float2*)(emb + (size_t)row * DIM))[lane];
            sum.x += e.x;       sum.y += e.y;
            sq.x  += e.x * e.x; sq.y  += e.y * e.y;
        }
    }
    // wave reduction of the linear-weight gather
    #pragma unroll
    for (int off = 16; off > 0; off >>= 1)
        wpart += __shfl_xor(wpart, off, 32);

    if (entity < n_entities) {
        float* rowp = outRows + (size_t)entity * (DIM + 2);
        ((float2*)rowp)[lane] = sum;                     // su -> cols [0,64)
        float2 bi;
        bi.x = 0.5f * (sum.x * sum.x - sq.x);
        bi.y = 0.5f * (sum.y * sum.y - sq.y);
        ((float2*)&lds_bi[wave][0])[lane] = bi;
        if (lane == 0) {
            lds_scal[wave] = wpart + (bias_ptr ? bias_ptr[0] : 0.f);
            rowp[ones_col] = 1.0f;
        }
    } else {
        ((float2*)&lds_bi[wave][0])[lane] = float2{0.f, 0.f};
        if (lane == 0) lds_scal[wave] = 0.f;
    }
    __syncthreads();

    // ---- wave 0: scal = bi @ h1 via V_WMMA_F32_16X16X4_F32 ----
    if (wave == 0) {
        const int m     = lane & 15;       // A-matrix row
        const int khalf = lane >> 4;       // 0: K={0,1}, 1: K={2,3}
        v8f acc = {};
        #pragma unroll
        for (int kb = 0; kb < 16; ++kb) {  // contract K=64 in steps of 4
            const int col = kb * 4 + khalf * 2;
            v2f a, b;
            a.x = lds_bi[m][col];
            a.y = lds_bi[m][col + 1];
            b.x = h1[col];                  // B[k,n] = h1[k] for every n
            b.y = h1[col + 1];
            acc = __builtin_amdgcn_wmma_f32_16x16x4_f32(
                false, a, false, b, (short)0, acc, false, false);
        }
        // C layout: acc[r] at lane L is row M = r + 8*(L>>4); all N equal.
        if (m == 0) {
            const int base = blockIdx.x * WAVES + 8 * khalf;
            #pragma unroll
            for (int r = 0; r < 8; ++r) {
                const int e = base + r;
                if (e < n_entities)
                    outRows[(size_t)e * (DIM + 2) + scal_col] =
                        acc[r] + lds_scal[8 * khalf + r];
            }
        }
    }
}

__global__ void enfm_copy_h2(const float* __restrict__ h2,
                             float* __restrict__ dst) {
    dst[threadIdx.x] = h2[threadIdx.x];
}

extern "C" void kernel_launch(void* const* d_in, const int* in_sizes, int n_in,
                              void* d_out, int out_size, void* d_ws, size_t ws_size,
                              hipStream_t stream) {
    const float* uemb  = (const float*)d_in[0];
    const float* iemb  = (const float*)d_in[1];
    const float* wu    = (const float*)d_in[2];
    const float* wi    = (const float*)d_in[3];
    const float* gbias = (const float*)d_in[4];
    const float* h1    = (const float*)d_in[5];
    const float* h2    = (const float*)d_in[6];
    const int*   uidx  = (const int*)d_in[7];
    const int*   iidx  = (const int*)d_in[8];

    const int B = in_sizes[7] / LFEAT;   // 16384
    const int I = in_sizes[8] / LFEAT;   // 16384

    float* P  = (float*)d_out;                 // [B, 66]
    float* Q  = P + (size_t)B * (DIM + 2);     // [I, 66]
    float* h2o = Q + (size_t)I * (DIM + 2);    // [64]

    dim3 block(512);
    // users: P = [su | scal(+bias) @ col 64 | 1 @ col 65]
    enfm_entity_kernel<<<dim3((B + WAVES - 1) / WAVES), block, 0, stream>>>(
        uemb, wu, h1, gbias, uidx, P, /*scal_col=*/DIM, /*ones_col=*/DIM + 1, B);
    // items: Q = [sv | 1 @ col 64 | scal @ col 65]
    enfm_entity_kernel<<<dim3((I + WAVES - 1) / WAVES), block, 0, stream>>>(
        iemb, wi, h1, nullptr, iidx, Q, /*scal_col=*/DIM + 1, /*ones_col=*/DIM, I);

    enfm_copy_h2<<<1, DIM, 0, stream>>>(h2, h2o);
}